// Net_4655744549304
// MI455X (gfx1250) — compile-verified
//
#include <hip/hip_runtime.h>

#define DIM 512
#define NN  128
#define EE  64
#define GG  16
#define VV  100000

typedef __attribute__((ext_vector_type(2))) float v2f;
typedef __attribute__((ext_vector_type(8))) float v8f;

// ---------------------------------------------------------------------------
// Kernel 1: base[n] = data_vecs[types[n], data[n]] @ data_weights[types[n]]
//                     + data_biases[types[n]]              (128 x 512 output)
// ---------------------------------------------------------------------------
__global__ __launch_bounds__(512) void base_kernel(
    const float* __restrict__ data_vecs,
    const float* __restrict__ data_weights,
    const float* __restrict__ data_biases,
    const int*   __restrict__ data,
    const int*   __restrict__ types,
    float* __restrict__ baseOut) {
  const int node = blockIdx.x;
  const int t = types[node];
  const long v = (long)data[node];
  const float* sel = data_vecs + ((long)t * VV + v) * DIM;
  __shared__ float s[DIM];
  for (int i = threadIdx.x; i < DIM; i += blockDim.x) s[i] = sel[i];
  __syncthreads();
  const float* W = data_weights + (size_t)t * DIM * DIM;
  const int f = threadIdx.x;  // blockDim == DIM
  float acc = data_biases[t * DIM + f];
#pragma unroll 4
  for (int k = 0; k < DIM; ++k) acc = fmaf(s[k], W[(size_t)k * DIM + f], acc);
  baseOut[node * DIM + f] = acc;
}

// ---------------------------------------------------------------------------
// WMMA f32 16x16x4 GEMM:  D[64,512] = A[64,512] @ B[512,512]   (row-major)
// 128 tiles of 16x16, 8 per wave (16 waves / block), K stepped by 4.
// Fragment layout per CDNA5 ISA 7.12.2 (32-bit A 16x4, B 4x16, C/D 16x16).
// ---------------------------------------------------------------------------
__device__ __forceinline__ void wmma_gemm_64x512(
    const float* __restrict__ A, const float* __restrict__ B,
    float* __restrict__ D, int tid) {
  const int wave = tid >> 5;
  const int lane = tid & 31;
  const int mn = lane & 15;   // M for A, N for B/D
  const int kh = lane >> 4;   // selects K pair {0,1} vs {2,3}
  for (int t = wave; t < 128; t += 16) {
    const int mt = t & 3;     // 0..3   (M tile)
    const int nt = t >> 2;    // 0..31  (N tile)
    v8f c = {0.f, 0.f, 0.f, 0.f, 0.f, 0.f, 0.f, 0.f};
    const float* Arow = A + (size_t)(mt * 16 + mn) * DIM;
    const float* Bcol = B + nt * 16 + mn;
#pragma unroll 4
    for (int k = 0; k < DIM; k += 4) {
      const int kk = k + 2 * kh;
      v2f a; a.x = Arow[kk];                 a.y = Arow[kk + 1];
      v2f b; b.x = Bcol[(size_t)kk * DIM];   b.y = Bcol[(size_t)(kk + 1) * DIM];
      c = __builtin_amdgcn_wmma_f32_16x16x4_f32(false, a, false, b,
                                                (short)0, c, false, false);
    }
#pragma unroll
    for (int j = 0; j < 8; ++j)
      D[(size_t)(mt * 16 + j + 8 * kh) * DIM + nt * 16 + mn] = c[j];
  }
}

// ---------------------------------------------------------------------------
// Kernel 2: one 512-thread workgroup per graph. Builds the tree in LDS,
// processes nodes bottom-up (level order). Plain nodes: block-wide GEMV.
// Ancestors of `pos` carry [64,512] embeddings: einsum (at pos) then
// WMMA GEMMs up the chain. Final score = (root contrib) @ score_weights.
// ---------------------------------------------------------------------------
__global__ __launch_bounds__(512) void graph_kernel(
    const float* __restrict__ baseEmb,
    const float* __restrict__ ew,      // [E, DIM, DIM]
    const float* __restrict__ eb,      // [E, DIM]
    const float* __restrict__ sw,      // [DIM]
    const int*   __restrict__ graphs,  // [G, N]
    const int*   __restrict__ edges,   // [N]
    const int*   __restrict__ posArr,  // [1]
    float* __restrict__ gbuf,          // per-graph scratch
    float* __restrict__ out) {         // [G, E]
  __shared__ int parents[NN];
  __shared__ int childStart[NN + 1];
  __shared__ int childList[NN];
  __shared__ int depthS[NN];
  __shared__ int isExp[NN];
  __shared__ int order[NN];
  __shared__ float vecs[DIM];
  __shared__ int meta[2];        // root, pos
  __shared__ float sDenomInv;

  const int g = blockIdx.x;
  const int tid = threadIdx.x;

  if (tid == 0) {
    for (int i = 0; i < NN; ++i) parents[i] = graphs[g * NN + i];
    // chain multiple roots (self-parents), last becomes the root
    int prevRoot = -1;
    for (int i = 0; i < NN; ++i)
      if (parents[i] == i) {
        if (prevRoot >= 0) parents[prevRoot] = i;
        prevRoot = i;
      }
    const int root = prevRoot;
    parents[root] = -1;
    // children CSR + #internal nodes
    for (int i = 0; i <= NN; ++i) childStart[i] = 0;
    for (int c = 0; c < NN; ++c) {
      const int p = parents[c];
      if (p >= 0) childStart[p + 1]++;
    }
    int nPar = 0;
    for (int i = 0; i < NN; ++i) if (childStart[i + 1] > 0) nPar++;
    for (int i = 0; i < NN; ++i) childStart[i + 1] += childStart[i];
    for (int i = 0; i < NN; ++i) order[i] = childStart[i];  // cursors
    for (int c = 0; c < NN; ++c) {
      const int p = parents[c];
      if (p >= 0) childList[order[p]++] = c;
    }
    // depths + level order (deepest first)
    int maxD = 0;
    for (int i = 0; i < NN; ++i) {
      int d = 0, a = i;
      while (parents[a] >= 0) { a = parents[a]; d++; }
      depthS[i] = d;
      if (d > maxD) maxD = d;
    }
    int o = 0;
    for (int lev = maxD; lev >= 0; --lev)
      for (int i = 0; i < NN; ++i)
        if (depthS[i] == lev) order[o++] = i;
    // mark strict ancestors of pos (they carry [E,DIM] embeddings)
    for (int i = 0; i < NN; ++i) isExp[i] = 0;
    const int pos = posArr[0];
    int a = parents[pos];
    while (a >= 0) { isExp[a] = 1; a = parents[a]; }
    meta[0] = root;
    meta[1] = pos;
    sDenomInv = 1.0f / (float)(nPar + 1);
  }
  __syncthreads();

  const int root = meta[0], pos = meta[1];
  const float denomInv = sDenomInv;

  float* embPlain = gbuf + (size_t)g * (NN * DIM + 2 * EE * DIM);
  float* embExp   = embPlain + NN * DIM;     // [E, DIM] current expanded emb
  float* tmp      = embExp + EE * DIM;       // [E, DIM] contribution scratch

  const int f = tid;  // this thread owns output dim f (blockDim == DIM)

  for (int oi = 0; oi < NN; ++oi) {
    const int idx = order[oi];
    const int cs = childStart[idx], ce = childStart[idx + 1];
    if (!isExp[idx]) {
      // ---- plain node (children are all plain; never contains pos's path)
      float acc = baseEmb[idx * DIM + f];
      for (int ci = cs; ci < ce; ++ci) {
        const int c = childList[ci];
        const int e = edges[c];
        __syncthreads();
        for (int i = tid; i < DIM; i += blockDim.x) vecs[i] = embPlain[c * DIM + i];
        __syncthreads();
        const float* W = ew + (size_t)e * DIM * DIM;
        float a2 = eb[e * DIM + f];
#pragma unroll 4
        for (int k = 0; k < DIM; ++k) a2 = fmaf(vecs[k], W[(size_t)k * DIM + f], a2);
        acc += a2;
      }
      if (ce > cs) acc = fmaxf(acc * denomInv, 0.0f);
      embPlain[idx * DIM + f] = acc;
      __syncthreads();
    } else {
      // ---- ancestor of pos: emb is [E, DIM]
      float v = baseEmb[idx * DIM + f];
      int expChild = -1;
      for (int ci = cs; ci < ce; ++ci) {
        const int c = childList[ci];
        if (c == pos || isExp[c]) {
          expChild = c;
          if (c != pos) v += eb[edges[c] * DIM + f];  // per-row bias folds into v
          continue;
        }
        const int e = edges[c];
        __syncthreads();
        for (int i = tid; i < DIM; i += blockDim.x) vecs[i] = embPlain[c * DIM + i];
        __syncthreads();
        const float* W = ew + (size_t)e * DIM * DIM;
        float a2 = eb[e * DIM + f];
#pragma unroll 4
        for (int k = 0; k < DIM; ++k) a2 = fmaf(vecs[k], W[(size_t)k * DIM + f], a2);
        v += a2;
      }
      __syncthreads();
      if (expChild == pos) {
        // einsum over ALL edge types: tmp[e][f] = pos_emb . EW[e][:,f] + eb[e][f]
        for (int i = tid; i < DIM; i += blockDim.x) vecs[i] = embPlain[pos * DIM + i];
        __syncthreads();
        for (int e = 0; e < EE; ++e) {
          const float* W = ew + (size_t)e * DIM * DIM;
          float a2 = eb[e * DIM + f];
#pragma unroll 4
          for (int k = 0; k < DIM; ++k) a2 = fmaf(vecs[k], W[(size_t)k * DIM + f], a2);
          tmp[e * DIM + f] = a2;
        }
      } else {
        // [64,512] @ [512,512] via v_wmma_f32_16x16x4_f32
        wmma_gemm_64x512(embExp, ew + (size_t)edges[expChild] * DIM * DIM, tmp, tid);
      }
      __syncthreads();
      for (int e = 0; e < EE; ++e) {
        const float val = tmp[e * DIM + f] + v;
        embExp[e * DIM + f] = fmaxf(val * denomInv, 0.0f);
      }
      __syncthreads();
    }
  }

  // ---- final contribution of the root over all E slots
  __syncthreads();
  if (pos == root) {
    // root stayed plain; expand over all edge types now
    for (int i = tid; i < DIM; i += blockDim.x) vecs[i] = embPlain[root * DIM + i];
    __syncthreads();
    for (int e = 0; e < EE; ++e) {
      const float* W = ew + (size_t)e * DIM * DIM;
      float a2 = eb[e * DIM + f];
#pragma unroll 4
      for (int k = 0; k < DIM; ++k) a2 = fmaf(vecs[k], W[(size_t)k * DIM + f], a2);
      tmp[e * DIM + f] = a2;
    }
  } else {
    wmma_gemm_64x512(embExp, ew + (size_t)edges[root] * DIM * DIM, tmp, tid);
  }
  __syncthreads();

  // ---- scores[g][e] = sum_f (tmp[e][f] + bias?) * sw[f]
  const int wave = tid >> 5, lane = tid & 31;
  const float* fbias = (pos == root) ? nullptr : (eb + (size_t)edges[root] * DIM);
  for (int e = wave; e < EE; e += 16) {
    float acc = 0.0f;
    for (int ff = lane; ff < DIM; ff += 32) {
      float val = tmp[e * DIM + ff];
      if (fbias) val += fbias[ff];
      acc += val * sw[ff];
    }
#pragma unroll
    for (int off = 16; off > 0; off >>= 1) acc += __shfl_down(acc, off, 32);
    if (lane == 0) out[g * EE + e] = acc;
  }
}

// ---------------------------------------------------------------------------
extern "C" void kernel_launch(void* const* d_in, const int* in_sizes, int n_in,
                              void* d_out, int out_size, void* d_ws, size_t ws_size,
                              hipStream_t stream) {
  const float* data_vecs    = (const float*)d_in[0];
  const float* data_weights = (const float*)d_in[1];
  const float* data_biases  = (const float*)d_in[2];
  const float* edge_weights = (const float*)d_in[3];
  const float* edge_biases  = (const float*)d_in[4];
  const float* score_w      = (const float*)d_in[5];
  const int*   data         = (const int*)d_in[6];
  const int*   types        = (const int*)d_in[7];
  const int*   graphs       = (const int*)d_in[8];
  const int*   edges        = (const int*)d_in[9];
  const int*   pos          = (const int*)d_in[10];

  // ws layout (floats): base[128*512] then per graph {embPlain, embExp, tmp}
  // total = 65536 + 16*131072 floats ~= 8.7 MB
  float* ws      = (float*)d_ws;
  float* baseBuf = ws;
  float* gbuf    = ws + NN * DIM;

  base_kernel<<<NN, DIM, 0, stream>>>(data_vecs, data_weights, data_biases,
                                      data, types, baseBuf);
  graph_kernel<<<GG, DIM, 0, stream>>>(baseBuf, edge_weights, edge_biases,
                                       score_w, graphs, edges, pos, gbuf,
                                       (float*)d_out);
}